// ScoreMatchingLoss_37847251812699
// MI455X (gfx1250) — compile-verified
//
#include <hip/hip_runtime.h>

#define G 4096
#define NBINS (G * 4)          // [class][group][{cnt,sum}] = 2*4096*2 floats = 64 KB
#define ACC_THREADS 256
#define ACC_BLOCKS 512

typedef __attribute__((ext_vector_type(2))) float v2f;
typedef __attribute__((ext_vector_type(8))) float v8f;

// ---------------------------------------------------------------------------
// Kernel 0: zero the global accumulator (harness does not re-poison d_ws).
// ---------------------------------------------------------------------------
__global__ void zero_acc_kernel(float* __restrict__ acc) {
    int i = blockIdx.x * blockDim.x + threadIdx.x;
    if (i < NBINS) acc[i] = 0.0f;
}

// ---------------------------------------------------------------------------
// Kernel 1: streaming segmented accumulation.
//   LDS layout: idx = (label << 13) + (group << 1); +0 = count, +1 = sum.
//   192 MB streamed once (HBM-bound, ~8.2 us floor at 23.3 TB/s);
//   scatter-adds hit a 64 KB LDS histogram, flushed once per block with
//   global f32 atomics into the L2-resident 64 KB accumulator.
// ---------------------------------------------------------------------------
__global__ __launch_bounds__(ACC_THREADS) void accum_kernel(
    const float* __restrict__ probs,
    const int* __restrict__ labels,
    const int* __restrict__ groups,
    float* __restrict__ acc,
    int nElem)
{
    __shared__ float lds[NBINS];   // 64 KB of the 320 KB WGP LDS
    for (int i = threadIdx.x; i < NBINS; i += blockDim.x) lds[i] = 0.0f;
    __syncthreads();

    const int nt  = gridDim.x * blockDim.x;
    const int tid = blockIdx.x * blockDim.x + threadIdx.x;
    const int nvec = nElem >> 2;

    const float4* __restrict__ p4 = (const float4*)probs;
    const int4*   __restrict__ l4 = (const int4*)labels;
    const int4*   __restrict__ g4 = (const int4*)groups;

    for (int v = tid; v < nvec; v += nt) {
        // lookahead prefetch of next grid-stride chunk (global_prefetch_b8;
        // speculative, OOB addresses silently dropped)
        __builtin_prefetch(&p4[v + nt], 0, 0);
        __builtin_prefetch(&l4[v + nt], 0, 0);
        __builtin_prefetch(&g4[v + nt], 0, 0);

        float4 p = p4[v];
        int4   l = l4[v];
        int4   g = g4[v];

        int i0 = (l.x << 13) + (g.x << 1);
        int i1 = (l.y << 13) + (g.y << 1);
        int i2 = (l.z << 13) + (g.z << 1);
        int i3 = (l.w << 13) + (g.w << 1);
        atomicAdd(&lds[i0], 1.0f); atomicAdd(&lds[i0 + 1], p.x);
        atomicAdd(&lds[i1], 1.0f); atomicAdd(&lds[i1 + 1], p.y);
        atomicAdd(&lds[i2], 1.0f); atomicAdd(&lds[i2 + 1], p.z);
        atomicAdd(&lds[i3], 1.0f); atomicAdd(&lds[i3 + 1], p.w);
    }
    // scalar tail (B = 16M is divisible by 4, but stay general)
    for (int i = (nvec << 2) + tid; i < nElem; i += nt) {
        float p = probs[i];
        int idx = (labels[i] << 13) + (groups[i] << 1);
        atomicAdd(&lds[idx], 1.0f); atomicAdd(&lds[idx + 1], p);
    }

    __syncthreads();
    for (int i = threadIdx.x; i < NBINS; i += blockDim.x)
        atomicAdd(&acc[i], lds[i]);
}

// ---------------------------------------------------------------------------
// Wave-wide (32-lane) sum via one V_WMMA_F32_16X16X4_F32.
// A (16x4 f32): lanes 0-15 hold {K0,K1}, lanes 16-31 hold {K2,K3}.
// With A[:,0]/A[:,2] = lane values, A[:,1]/A[:,3] = 0 and B = ones:
//   D[m][n] = v(m) + v(m+16).
// Lane L<16 holds D[0..7][L] in its 8 C/D VGPRs; lane L>=16 holds D[8..15][*].
// Sum the 8 regs, then one xor-16 shuffle gives the full 32-lane sum in every
// lane. Requires EXEC all-ones: callers invoke uniformly from all 256 threads.
// ---------------------------------------------------------------------------
__device__ __forceinline__ float wave_sum32(float v) {
    v2f a; a[0] = v;     a[1] = 0.0f;
    v2f b; b[0] = 1.0f;  b[1] = 1.0f;
    v8f c = {};
    c = __builtin_amdgcn_wmma_f32_16x16x4_f32(
            /*neg_a=*/false, a, /*neg_b=*/false, b,
            /*c_mod=*/(short)0, c, /*reuse_a=*/false, /*reuse_b=*/false);
    float s = c[0] + c[1] + c[2] + c[3] + c[4] + c[5] + c[6] + c[7];
    s += __shfl_xor(s, 16, 32);
    return s;
}

__device__ __forceinline__ float block_sum256(float v, float* red, int tid) {
    float ws = wave_sum32(v);
    int wave = tid >> 5;
    if ((tid & 31) == 0) red[wave] = ws;
    __syncthreads();
    float total = red[0] + red[1] + red[2] + red[3] +
                  red[4] + red[5] + red[6] + red[7];
    __syncthreads();
    return total;     // in every thread
}

// ---------------------------------------------------------------------------
// Kernel 2: O(G) epilogue — per-class variance of group means, final select.
// acc layout: class0 (label==0) at [0..8191], class1 at [8192..16383];
//             per group: +0 = cnt, +1 = sum.
// ---------------------------------------------------------------------------
__global__ __launch_bounds__(256) void finalize_kernel(
    const float* __restrict__ acc, float* __restrict__ out)
{
    __shared__ float red[8];
    const int tid = threadIdx.x;

    // pass 1: n_valid and sum of means per class
    float n_pos = 0.0f, sm_pos = 0.0f, n_neg = 0.0f, sm_neg = 0.0f;
    for (int g = tid; g < G; g += 256) {
        float cn = acc[2 * g],            sn = acc[2 * g + 1];
        float cp = acc[2 * G * 2 + 2 * g], sp = acc[2 * G * 2 + 2 * g + 1];
        float vp = (cp >= 1.0f) ? 1.0f : 0.0f;
        float vn = (cn >= 1.0f) ? 1.0f : 0.0f;
        n_pos += vp;  sm_pos += vp * (sp / fmaxf(cp, 1.0f));
        n_neg += vn;  sm_neg += vn * (sn / fmaxf(cn, 1.0f));
    }
    float N_pos = block_sum256(n_pos, red, tid);
    float S_pos = block_sum256(sm_pos, red, tid);
    float N_neg = block_sum256(n_neg, red, tid);
    float S_neg = block_sum256(sm_neg, red, tid);

    float mom_pos = S_pos / fmaxf(N_pos, 1.0f);
    float mom_neg = S_neg / fmaxf(N_neg, 1.0f);

    // pass 2: sum of squared deviations of valid group means
    float vp_acc = 0.0f, vn_acc = 0.0f;
    for (int g = tid; g < G; g += 256) {
        float cn = acc[2 * g],            sn = acc[2 * g + 1];
        float cp = acc[2 * G * 2 + 2 * g], sp = acc[2 * G * 2 + 2 * g + 1];
        float vp = (cp >= 1.0f) ? 1.0f : 0.0f;
        float vn = (cn >= 1.0f) ? 1.0f : 0.0f;
        float mp = sp / fmaxf(cp, 1.0f);
        float mn = sn / fmaxf(cn, 1.0f);
        float dp = mp - mom_pos;
        float dn = mn - mom_neg;
        vp_acc += vp * dp * dp;
        vn_acc += vn * dn * dn;
    }
    float Vp = block_sum256(vp_acc, red, tid);
    float Vn = block_sum256(vn_acc, red, tid);

    if (tid == 0) {
        float var_pos = Vp / fmaxf(N_pos - 1.0f, 1.0f);
        float var_neg = Vn / fmaxf(N_neg - 1.0f, 1.0f);
        bool pos_ok = N_pos >= 2.0f;
        bool neg_ok = N_neg >= 2.0f;
        float loss = (pos_ok && neg_ok) ? 0.5f * (var_pos + var_neg)
                   : (pos_ok ? var_pos : (neg_ok ? var_neg : 0.0f));
        out[0] = loss;
    }
}

// ---------------------------------------------------------------------------
extern "C" void kernel_launch(void* const* d_in, const int* in_sizes, int n_in,
                              void* d_out, int out_size, void* d_ws, size_t ws_size,
                              hipStream_t stream) {
    const float* probs  = (const float*)d_in[0];
    const int*   labels = (const int*)d_in[1];
    const int*   groups = (const int*)d_in[2];
    float* acc = (float*)d_ws;     // needs NBINS * 4 = 64 KB
    float* out = (float*)d_out;
    const int B = in_sizes[0];

    zero_acc_kernel<<<(NBINS + 255) / 256, 256, 0, stream>>>(acc);
    accum_kernel<<<ACC_BLOCKS, ACC_THREADS, 0, stream>>>(probs, labels, groups, acc, B);
    finalize_kernel<<<1, 256, 0, stream>>>(acc, out);
}